// EmformerAttention_48189533061523
// MI455X (gfx1250) — compile-verified
//
#include <hip/hip_runtime.h>
#include <cmath>

// ---------------------------------------------------------------------------
// Emformer attention for MI455X (gfx1250, wave32).
// fp32 end-to-end via V_WMMA_F32_16X16X4_F32 (CDNA5 fp32 matrix path).
// Projection GEMMs stage the shared 16-row A strip in LDS with the Tensor
// Data Mover (tensor_load_to_lds, TENSORcnt), double-buffered over K chunks.
// ---------------------------------------------------------------------------

typedef float v2f __attribute__((ext_vector_type(2)));
typedef float v8f __attribute__((ext_vector_type(8)));
typedef unsigned int u32x4 __attribute__((ext_vector_type(4)));
typedef int i32x4 __attribute__((ext_vector_type(4)));
typedef int i32x8 __attribute__((ext_vector_type(8)));

#define D_EMB    1024
#define N_HEAD   16
#define HEAD_DIM 64
#define U_LEN    2048
#define R_LEN    256
#define L_LEN    1024
#define M_LEN    512
#define Q_LEN    (U_LEN + R_LEN)                  // 2304
#define KV_LEN   (M_LEN + R_LEN + L_LEN + U_LEN)  // 3840

#define KCHUNK    256                 // K elements staged per TDM transfer
#define APITCH    260                 // LDS row pitch in dwords (256 + 4 pad)
#define NCHUNKS   (D_EMB / KCHUNK)    // 4

// ---- TDM: DMA a 2D fp32 tile (rows x cols, row stride in dwords) to LDS ---
// D# per CDNA5 ISA ch.8: group0 = {flags, lds_addr, global_addr, type=2},
// group1 = {data_size=4B | pad(interval=256dw, amount=4dw), dims/strides}.
// Padding gives LDS row pitch 260 dwords -> bank-conflict-free, 8B aligned.
__device__ __forceinline__ void tdm_load_2d(unsigned lds_byte_off,
                                            const float* gsrc,
                                            unsigned cols_dw, unsigned rows,
                                            unsigned row_stride_dw) {
  unsigned long long ga = (unsigned long long)(uintptr_t)gsrc;
  u32x4 g0;
  g0[0] = 1u;                                        // count=1 (user mode)
  g0[1] = lds_byte_off;                              // lds_addr [63:32]
  g0[2] = (unsigned)(ga & 0xFFFFFFFFu);              // global_addr[31:0]
  g0[3] = (unsigned)((ga >> 32) & 0x1FFFFFFu)        // global_addr[56:32]
          | (2u << 30);                              // type=2 ("image")
  i32x8 g1;
  g1[0] = (int)((2u << 16)                           // data_size = 4B
                | (1u << 20)                         // pad_enable
                | (7u << 22)                         // pad_interval = 256 dw
                | (3u << 25));                       // pad_amount  = 4 dw
  g1[1] = (int)((cols_dw & 0xFFFFu) << 16);          // tensor_dim0[15:0]
  g1[2] = (int)(((cols_dw >> 16) & 0xFFFFu)          // tensor_dim0[31:16]
                | ((rows & 0xFFFFu) << 16));         // tensor_dim1[15:0]
  g1[3] = (int)((cols_dw & 0xFFFFu) << 16);          // tile_dim0
  g1[4] = (int)(rows & 0xFFFFu);                     // tile_dim1 (tile_dim2=0)
  g1[5] = (int)row_stride_dw;                        // tensor_dim0_stride lo
  g1[6] = 0;
  g1[7] = 0;
  i32x4 z4 = {0, 0, 0, 0};
#if __clang_major__ >= 23
  i32x8 z8 = {0, 0, 0, 0, 0, 0, 0, 0};
  __builtin_amdgcn_tensor_load_to_lds(g0, g1, z4, z4, z8, 0);
#else
  __builtin_amdgcn_tensor_load_to_lds(g0, g1, z4, z4, 0);
#endif
}

// ---- generic block body: 16-row A strip (LDS via TDM) x 512 cols ----------
// A frag: lane(l16,half) holds A[m=l16][k+2*half+{0,1}]   (ISA 16x4 layout)
// B frag: lane(l16,half) holds W[n=l16][k+2*half+{0,1}]
// C frag: vgpr j, half h -> (M = j+8h, N = l16)
__device__ __forceinline__ void gemm_strip(float atile[2][16 * APITCH],
                                           const float* aseg,   // strip row 0
                                           const float* wBase,  // W + 2*half
                                           int nbase, int l16, int half, int wv,
                                           v8f acc[4]) {
  if (wv == 0) tdm_load_2d((unsigned)(uintptr_t)(void*)&atile[0][0],
                           aseg, KCHUNK, 16, D_EMB);
  for (int c = 0; c < NCHUNKS; ++c) {
    if (wv == 0) {
      if (c + 1 < NCHUNKS) {
        tdm_load_2d((unsigned)(uintptr_t)(void*)&atile[(c + 1) & 1][0],
                    aseg + (c + 1) * KCHUNK, KCHUNK, 16, D_EMB);
        __builtin_amdgcn_s_wait_tensorcnt(1);   // chunk c landed
      } else {
        __builtin_amdgcn_s_wait_tensorcnt(0);
      }
    }
    __syncthreads();                            // chunk c visible to block
    const float* ab = &atile[c & 1][l16 * APITCH + 2 * half];
    const float* wb = wBase + c * KCHUNK;
#pragma unroll 4
    for (int k = 0; k < KCHUNK; k += 4) {
      v2f a = *(const v2f*)(ab + k);
#pragma unroll
      for (int t = 0; t < 4; ++t) {
        v2f b = *(const v2f*)(wb + (size_t)t * 16 * D_EMB + k);
        acc[t] = __builtin_amdgcn_wmma_f32_16x16x4_f32(false, a, false, b,
                                                       (short)0, acc[t],
                                                       false, false);
      }
    }
    __syncthreads();                            // done reading buf c&1
  }
  (void)nbase;
}

__device__ __forceinline__ float rowmax16(float x) {
  x = fmaxf(x, __shfl_xor(x, 1, 32));
  x = fmaxf(x, __shfl_xor(x, 2, 32));
  x = fmaxf(x, __shfl_xor(x, 4, 32));
  x = fmaxf(x, __shfl_xor(x, 8, 32));
  return x;
}
__device__ __forceinline__ float rowsum16(float x) {
  x += __shfl_xor(x, 1, 32);
  x += __shfl_xor(x, 2, 32);
  x += __shfl_xor(x, 4, 32);
  x += __shfl_xor(x, 8, 32);
  return x;
}

// ---- kv = concat(mem,rctx,utt) @ Wkv.T + bkv -> scatter into key/value ----
// grid: 176 strips x 4 col-blocks (512 cols each); wave = 16x64 outputs.
__global__ __launch_bounds__(256) void kv_gemm_kernel(
    const float* __restrict__ mem,  const float* __restrict__ rctx,
    const float* __restrict__ utt,  const float* __restrict__ Wkv,
    const float* __restrict__ bkv,  float* __restrict__ keyOut,
    float* __restrict__ valOut) {
  __shared__ float atile[2][16 * APITCH];
  const int lane = threadIdx.x & 31, half = lane >> 4, l16 = lane & 15;
  const int wv = threadIdx.x >> 5;
  const int tm = blockIdx.x >> 2;
  const int nbase = (blockIdx.x & 3) * 512 + wv * 64;

  const int r0 = tm * 16;                 // strip never crosses segment bounds
  const float* aseg;
  if (r0 < M_LEN)               aseg = mem  + (size_t)r0 * D_EMB;
  else if (r0 < M_LEN + R_LEN)  aseg = rctx + (size_t)(r0 - M_LEN) * D_EMB;
  else                          aseg = utt  + (size_t)(r0 - M_LEN - R_LEN) * D_EMB;

  const float* wBase = Wkv + (size_t)(nbase + l16) * D_EMB + 2 * half;
  v8f acc[4] = {};
  gemm_strip(atile, aseg, wBase, nbase, l16, half, wv, acc);

#pragma unroll
  for (int t = 0; t < 4; ++t) {
    const int n = nbase + t * 16 + l16;
    const float bias = bkv[n];
    float* dst; int col;
    if (n < D_EMB) { dst = keyOut; col = n; }
    else           { dst = valOut; col = n - D_EMB; }
#pragma unroll
    for (int j = 0; j < 8; ++j) {
      const int row  = tm * 16 + j + 8 * half;
      const int orow = (row < M_LEN + R_LEN) ? row : row + L_LEN;
      dst[(size_t)orow * D_EMB + col] = acc[t][j] + bias;
    }
  }
}

// ---- q = concat(rctx,utt) @ Wq.T + bq, pre-scaled by 1/sqrt(64) -----------
// grid: 144 strips x 2 col-blocks
__global__ __launch_bounds__(256) void q_gemm_kernel(
    const float* __restrict__ rctx, const float* __restrict__ utt,
    const float* __restrict__ Wq,   const float* __restrict__ bq,
    float* __restrict__ qOut) {
  __shared__ float atile[2][16 * APITCH];
  const int lane = threadIdx.x & 31, half = lane >> 4, l16 = lane & 15;
  const int wv = threadIdx.x >> 5;
  const int tm = blockIdx.x >> 1;
  const int nbase = (blockIdx.x & 1) * 512 + wv * 64;

  const int r0 = tm * 16;
  const float* aseg = (r0 < R_LEN) ? rctx + (size_t)r0 * D_EMB
                                   : utt  + (size_t)(r0 - R_LEN) * D_EMB;
  const float* wBase = Wq + (size_t)(nbase + l16) * D_EMB + 2 * half;
  v8f acc[4] = {};
  gemm_strip(atile, aseg, wBase, nbase, l16, half, wv, acc);

  const float scaling = 0.125f;           // HEAD_DIM^-0.5
#pragma unroll
  for (int t = 0; t < 4; ++t) {
    const int n = nbase + t * 16 + l16;
    const float bias = bq[n];
#pragma unroll
    for (int j = 0; j < 8; ++j) {
      const int row = tm * 16 + j + 8 * half;
      qOut[(size_t)row * D_EMB + n] = (acc[t][j] + bias) * scaling;
    }
  }
}

// ---- flash attention: one wave = one (head, 16-row q tile) ----------------
__global__ __launch_bounds__(256) void attn_kernel(
    const float* __restrict__ q,   const float* __restrict__ key,
    const float* __restrict__ val, float* __restrict__ attn) {
  __shared__ float pbuf[8][16][20];       // per-wave P tile, padded rows
  const int lane = threadIdx.x & 31, half = lane >> 4, l16 = lane & 15;
  const int w = threadIdx.x >> 5;
  const int tile = blockIdx.x * 8 + w;
  const int h  = tile / (Q_LEN / 16);
  const int qt = tile - h * (Q_LEN / 16);
  const int q0 = qt * 16;

  // hoisted Q fragments (loop-invariant over all KV chunks)
  const float* qRow = q + (size_t)(q0 + l16) * D_EMB + h * HEAD_DIM + 2 * half;
  v2f qa[16];
#pragma unroll
  for (int d = 0; d < 16; ++d) qa[d] = *(const v2f*)(qRow + 4 * d);

  float mrow[8], lrow[8];
  v8f acc[4] = {};
#pragma unroll
  for (int j = 0; j < 8; ++j) { mrow[j] = -INFINITY; lrow[j] = 0.0f; }

  for (int kv0 = 0; kv0 < KV_LEN; kv0 += 16) {
    // S = (q*scale) @ K_chunk^T : 16 wmma over d=64
    const float* kRow = key + (size_t)(kv0 + l16) * D_EMB + h * HEAD_DIM + 2 * half;
    v8f s = {};
#pragma unroll
    for (int d = 0; d < 16; ++d) {
      v2f b = *(const v2f*)(kRow + 4 * d);
      s = __builtin_amdgcn_wmma_f32_16x16x4_f32(false, qa[d], false, b,
                                                (short)0, s, false, false);
    }
    // online softmax; row = j + 8*half, column = l16
#pragma unroll
    for (int j = 0; j < 8; ++j) {
      const float sv    = s[j];
      const float rm    = rowmax16(sv);
      const float mnew  = fmaxf(mrow[j], rm);
      const float alpha = __expf(mrow[j] - mnew);
      const float p     = __expf(sv - mnew);
      const float psum  = rowsum16(p);
      lrow[j] = lrow[j] * alpha + psum;
      mrow[j] = mnew;
#pragma unroll
      for (int t = 0; t < 4; ++t) acc[t][j] *= alpha;
      pbuf[w][j + 8 * half][l16] = p;     // same-wave LDS: in-order, no barrier
    }
    // O += P @ V_chunk : K=16 -> 4 wmma per 16-wide N tile, 4 tiles
#pragma unroll
    for (int kk = 0; kk < 4; ++kk) {
      v2f pa = *(const v2f*)(&pbuf[w][l16][4 * kk + 2 * half]);
      const float* vp = val + (size_t)(kv0 + 4 * kk + 2 * half) * D_EMB
                            + h * HEAD_DIM + l16;
#pragma unroll
      for (int t = 0; t < 4; ++t) {
        v2f vb = { vp[t * 16], vp[t * 16 + D_EMB] };
        acc[t] = __builtin_amdgcn_wmma_f32_16x16x4_f32(false, pa, false, vb,
                                                       (short)0, acc[t],
                                                       false, false);
      }
    }
  }
#pragma unroll
  for (int t = 0; t < 4; ++t)
#pragma unroll
    for (int j = 0; j < 8; ++j) {
      const int row = q0 + j + 8 * half;
      attn[(size_t)row * D_EMB + h * HEAD_DIM + t * 16 + l16] = acc[t][j] / lrow[j];
    }
}

// ---- out = attn @ Wo.T + bo ----------------------------------------------
// grid: 144 strips x 2 col-blocks
__global__ __launch_bounds__(256) void out_gemm_kernel(
    const float* __restrict__ attn, const float* __restrict__ Wo,
    const float* __restrict__ bo,   float* __restrict__ out) {
  __shared__ float atile[2][16 * APITCH];
  const int lane = threadIdx.x & 31, half = lane >> 4, l16 = lane & 15;
  const int wv = threadIdx.x >> 5;
  const int tm = blockIdx.x >> 1;
  const int nbase = (blockIdx.x & 1) * 512 + wv * 64;

  const float* aseg = attn + (size_t)(tm * 16) * D_EMB;
  const float* wBase = Wo + (size_t)(nbase + l16) * D_EMB + 2 * half;
  v8f acc[4] = {};
  gemm_strip(atile, aseg, wBase, nbase, l16, half, wv, acc);

#pragma unroll
  for (int t = 0; t < 4; ++t) {
    const int n = nbase + t * 16 + l16;
    const float bias = bo[n];
#pragma unroll
    for (int j = 0; j < 8; ++j) {
      const int row = tm * 16 + j + 8 * half;
      out[(size_t)row * D_EMB + n] = acc[t][j] + bias;
    }
  }
}

extern "C" void kernel_launch(void* const* d_in, const int* in_sizes, int n_in,
                              void* d_out, int out_size, void* d_ws, size_t ws_size,
                              hipStream_t stream) {
  const float* utt  = (const float*)d_in[0];
  const float* rctx = (const float*)d_in[1];
  const float* mem  = (const float*)d_in[2];
  const float* lck  = (const float*)d_in[3];
  const float* lcv  = (const float*)d_in[4];
  const float* Wq   = (const float*)d_in[5];
  const float* bq   = (const float*)d_in[6];
  const float* Wkv  = (const float*)d_in[7];
  const float* bkv  = (const float*)d_in[8];
  const float* Wo   = (const float*)d_in[9];
  const float* bo   = (const float*)d_in[10];

  float* out    = (float*)d_out;                       // (2304,1024)
  float* keyOut = out + (size_t)Q_LEN * D_EMB;         // (3840,1024)
  float* valOut = keyOut + (size_t)KV_LEN * D_EMB;     // (3840,1024)

  float* qBuf    = (float*)d_ws;                       // (2304,1024) scaled q
  float* attnBuf = qBuf + (size_t)Q_LEN * D_EMB;       // (2304,1024)

  // left-context rows straight into key/value outputs (graph-capturable d2d)
  hipMemcpyAsync(keyOut + (size_t)(M_LEN + R_LEN) * D_EMB, lck,
                 (size_t)L_LEN * D_EMB * sizeof(float),
                 hipMemcpyDeviceToDevice, stream);
  hipMemcpyAsync(valOut + (size_t)(M_LEN + R_LEN) * D_EMB, lcv,
                 (size_t)L_LEN * D_EMB * sizeof(float),
                 hipMemcpyDeviceToDevice, stream);

  kv_gemm_kernel<<<704, 256, 0, stream>>>(mem, rctx, utt, Wkv, bkv, keyOut, valOut);
  q_gemm_kernel<<<288, 256, 0, stream>>>(rctx, utt, Wq, bq, qBuf);
  attn_kernel<<<288, 256, 0, stream>>>(qBuf, keyOut, valOut, attnBuf);
  out_gemm_kernel<<<288, 256, 0, stream>>>(attnBuf, Wo, bo, out);
}